// Gemma3nAltUp_19026705121872
// MI455X (gfx1250) — compile-verified
//
#include <hip/hip_runtime.h>

#define T_TOK 8192
#define H_DIM 2048
#define N_IN  4
#define EPSF  1e-6f

typedef float v2f __attribute__((ext_vector_type(2)));
typedef float v8f __attribute__((ext_vector_type(8)));

// ---------------------------------------------------------------------------
// Bpre[n][k] = router_norm_w[k] * modality_router_w[n][k]  (n<4), else 0.
// Folds the rmsnorm weight into the router GEMM's B operand (padded to 16 cols).
// ---------------------------------------------------------------------------
__global__ __launch_bounds__(256) void init_bpre_kernel(
    const float* __restrict__ norm_w, const float* __restrict__ router_w,
    float* __restrict__ bpre)
{
    int idx = blockIdx.x * blockDim.x + threadIdx.x;   // 16 * 2048
    int n = idx >> 11;
    int k = idx & (H_DIM - 1);
    bpre[idx] = (n < N_IN) ? norm_w[k] * router_w[n * H_DIM + k] : 0.0f;
}

// ---------------------------------------------------------------------------
// Per 16-token tile: D[t][n] = sum_k X[t][k]*Bpre[n][k] via V_WMMA_F32_16X16X4_F32,
// sumsq via VALU, then m = tanh(D * rsqrt(mean+eps)/H), then coefs = m @ Pw^T + bias.
// 4 waves per block, K split across waves, LDS reduction.
// ---------------------------------------------------------------------------
__global__ __launch_bounds__(128) void router_coef_kernel(
    const float* __restrict__ X,     // [T, H] (hidden_states[0] or activated)
    const float* __restrict__ Bpre,  // [16, H]
    const float* __restrict__ Pw,    // [NC, 4]
    float* __restrict__ coef_out,    // [T, NC]
    int NC, float bias)
{
    __shared__ float Cred[4][16][16];
    __shared__ float ssred[4][16];
    __shared__ float s_lds[16];
    __shared__ float m_lds[16][16];

    const int tid   = threadIdx.x;
    const int wave  = tid >> 5;
    const int lane  = tid & 31;
    const int tok   = lane & 15;             // token row (A) / col n (B)
    const int krow  = (lane >> 4) << 1;      // 0 (lanes 0-15) or 2 (lanes 16-31)
    const int mbase = (lane >> 4) << 3;      // C rows: 0..7 or 8..15

    const int KSEG = H_DIM / 4;              // 512 K per wave
    const float* xrow = X    + (size_t)(blockIdx.x * 16 + tok) * H_DIM + (size_t)wave * KSEG + krow;
    const float* brow = Bpre + (size_t)tok * H_DIM                     + (size_t)wave * KSEG + krow;

    v8f C = {};
    float ss = 0.0f;
    #pragma unroll 4
    for (int k = 0; k < KSEG; k += 4) {
        v2f a = *(const v2f*)(xrow + k);     // A 16x4 f32 tile slice
        v2f b = *(const v2f*)(brow + k);     // B 4x16 f32 tile slice
        C = __builtin_amdgcn_wmma_f32_16x16x4_f32(
                /*neg_a=*/false, a, /*neg_b=*/false, b,
                /*c_mod=*/(short)0, C, /*reuse_a=*/false, /*reuse_b=*/false);
        ss = fmaf(a.x, a.x, ss);
        ss = fmaf(a.y, a.y, ss);
    }
    // lanes L and L+16 together cover all K of token L -> combine halves
    ss += __shfl_xor(ss, 16);

    #pragma unroll
    for (int v = 0; v < 8; ++v) Cred[wave][mbase + v][tok] = C[v];
    if (lane < 16) ssred[wave][lane] = ss;
    __syncthreads();

    if (tid < 16) {
        float sst = ssred[0][tid] + ssred[1][tid] + ssred[2][tid] + ssred[3][tid];
        s_lds[tid] = rsqrtf(sst * (1.0f / H_DIM) + EPSF) * (1.0f / H_DIM);
    }
    __syncthreads();

    if (wave == 0) {
        #pragma unroll
        for (int v = 0; v < 8; ++v) {
            int m = mbase + v;
            float d = Cred[0][m][tok] + Cred[1][m][tok] + Cred[2][m][tok] + Cred[3][m][tok];
            m_lds[m][tok] = tanhf(d * s_lds[m]);
        }
    }
    __syncthreads();

    if (tid < 16) {
        float m0 = m_lds[tid][0], m1 = m_lds[tid][1];
        float m2 = m_lds[tid][2], m3 = m_lds[tid][3];
        float* out = coef_out + (size_t)(blockIdx.x * 16 + tid) * NC;
        for (int j = 0; j < NC; ++j) {
            const float* p = Pw + j * 4;
            out[j] = fmaf(m0, p[0], fmaf(m1, p[1], fmaf(m2, p[2], fmaf(m3, p[3], bias))));
        }
    }
}

// ---------------------------------------------------------------------------
// Fused streaming pass: one block per token, float4 per thread across H.
//   pred[m]  = hs[m] + sum_n coef[m*4+n] * hs[n]
//   innov    = act - pred[0]
//   out[m]   = innov * ccoef[m] + pred[m]
// ---------------------------------------------------------------------------
__global__ __launch_bounds__(512) void altup_fuse_kernel(
    const float4* __restrict__ hs,    // [4, T, H/4]
    const float4* __restrict__ act,   // [T, H/4]
    const float*  __restrict__ pcoef, // [T, 16]
    const float*  __restrict__ ccoef, // [T, 4]
    float4* __restrict__ out)         // [4, T, H/4]
{
    const int HV = H_DIM / 4;                       // 512
    const size_t PL = (size_t)T_TOK * HV;
    int t  = blockIdx.x;
    size_t base = (size_t)t * HV + threadIdx.x;

    float4 h[4];
    h[0] = hs[base];
    h[1] = hs[PL + base];
    h[2] = hs[2 * PL + base];
    h[3] = hs[3 * PL + base];
    float4 a = act[base];

    float c[16];
    #pragma unroll
    for (int j = 0; j < 16; ++j) c[j] = pcoef[(size_t)t * 16 + j];   // uniform -> s_load
    float cc[4];
    #pragma unroll
    for (int j = 0; j < 4; ++j) cc[j] = ccoef[(size_t)t * 4 + j];

    float4 p[4];
    #pragma unroll
    for (int m = 0; m < 4; ++m) {
        float4 r = h[m];
        #pragma unroll
        for (int n = 0; n < 4; ++n) {
            float cn = c[m * 4 + n];
            r.x = fmaf(cn, h[n].x, r.x);
            r.y = fmaf(cn, h[n].y, r.y);
            r.z = fmaf(cn, h[n].z, r.z);
            r.w = fmaf(cn, h[n].w, r.w);
        }
        p[m] = r;
    }

    float4 innov;
    innov.x = a.x - p[0].x;
    innov.y = a.y - p[0].y;
    innov.z = a.z - p[0].z;
    innov.w = a.w - p[0].w;

    #pragma unroll
    for (int m = 0; m < 4; ++m) {
        float4 o;
        o.x = fmaf(innov.x, cc[m], p[m].x);
        o.y = fmaf(innov.y, cc[m], p[m].y);
        o.z = fmaf(innov.z, cc[m], p[m].z);
        o.w = fmaf(innov.w, cc[m], p[m].w);
        out[(size_t)m * PL + base] = o;
    }
}

// ---------------------------------------------------------------------------
extern "C" void kernel_launch(void* const* d_in, const int* in_sizes, int n_in,
                              void* d_out, int out_size, void* d_ws, size_t ws_size,
                              hipStream_t stream) {
    const float* hs  = (const float*)d_in[0];   // [4, T, H]
    const float* act = (const float*)d_in[1];   // [T, H]
    const float* nw  = (const float*)d_in[2];   // [H]
    const float* rw  = (const float*)d_in[3];   // [4, H]
    const float* pw  = (const float*)d_in[4];   // [16, 4]
    const float* cw  = (const float*)d_in[5];   // [4, 4]
    float* out = (float*)d_out;

    float* bpre  = (float*)d_ws;                         // 16*H      = 128 KB
    float* pcoef = bpre + 16 * H_DIM;                    // T*16      = 512 KB
    float* ccoef = pcoef + (size_t)T_TOK * 16;           // T*4       = 128 KB

    init_bpre_kernel<<<(16 * H_DIM) / 256, 256, 0, stream>>>(nw, rw, bpre);
    // hidden_states[ACTIVE_IDX=0] is just the base pointer
    router_coef_kernel<<<T_TOK / 16, 128, 0, stream>>>(hs,  bpre, pw, pcoef, 16, 0.0f);
    router_coef_kernel<<<T_TOK / 16, 128, 0, stream>>>(act, bpre, cw, ccoef, 4, 1.0f);
    altup_fuse_kernel<<<T_TOK, 512, 0, stream>>>((const float4*)hs, (const float4*)act,
                                                 pcoef, ccoef, (float4*)out);
}